// FastSpeech2Loss_59442347377564
// MI455X (gfx1250) — compile-verified
//
#include <hip/hip_runtime.h>

// ---------------- problem constants ----------------
#define KMIX 5
#define B_ 16
#define T_ 864
#define M_ 80
#define S_ 128
#define NBT (B_ * T_)             // 13824
#define NBS (B_ * S_)             // 2048
#define NCELL (B_ * T_ * M_)      // 1,105,920 cells, 200 B each
#define CELL_BYTES 200
#define TILE_CELLS 128
#define TILE_BYTES (TILE_CELLS * CELL_BYTES)   // 25600 B
#define NTILES (NCELL / TILE_CELLS)            // 8640 (exact)
#define VL_BLOCKS 1080                         // 8 tiles per block
#define VL_THREADS 128                         // 4 waves
#define WAVE_REGION (TILE_BYTES / 4)           // 6400 B per wave
#define ASYNC_PER_WAVE 13                      // 12 x b128 + 1 x b64
#define LOG_2PI 1.8378770664093453f
#define MIN_VAR 1e-4f

// ---------------- CDNA5 async-to-LDS path ----------------
#if defined(__has_builtin)
#if __has_builtin(__builtin_amdgcn_global_load_async_to_lds_b128)
#define HAVE_ASYNC_BUILTIN 1
#endif
#if __has_builtin(__builtin_amdgcn_s_wait_asynccnt)
#define HAVE_WAIT_BUILTIN 1
#endif
#endif

typedef int v4i_t __attribute__((vector_size(16)));
typedef int v2i_t __attribute__((vector_size(8)));
typedef __attribute__((address_space(1))) v4i_t* gp_v4i;
typedef __attribute__((address_space(3))) v4i_t* lp_v4i;
typedef __attribute__((address_space(1))) v2i_t* gp_v2i;
typedef __attribute__((address_space(3))) v2i_t* lp_v2i;

__device__ __forceinline__ void async_copy_b128(const void* g, void* l) {
#ifdef HAVE_ASYNC_BUILTIN
  __builtin_amdgcn_global_load_async_to_lds_b128((gp_v4i)g, (lp_v4i)l, 0, 0);
#else
  unsigned lds_addr = (unsigned)(unsigned long long)l;  // low 32b of generic LDS ptr
  asm volatile("global_load_async_to_lds_b128 %0, %1, off"
               :: "v"(lds_addr), "v"(g) : "memory");
#endif
}

__device__ __forceinline__ void async_copy_b64(const void* g, void* l) {
#ifdef HAVE_ASYNC_BUILTIN
  __builtin_amdgcn_global_load_async_to_lds_b64((gp_v2i)g, (lp_v2i)l, 0, 0);
#else
  unsigned lds_addr = (unsigned)(unsigned long long)l;
  asm volatile("global_load_async_to_lds_b64 %0, %1, off"
               :: "v"(lds_addr), "v"(g) : "memory");
#endif
}

__device__ __forceinline__ void wait_async_prev() {  // allow next tile in flight
#ifdef HAVE_WAIT_BUILTIN
  __builtin_amdgcn_s_wait_asynccnt(ASYNC_PER_WAVE);
#else
  asm volatile("s_wait_asynccnt 13" ::: "memory");
#endif
}

__device__ __forceinline__ void wait_async_all() {
#ifdef HAVE_WAIT_BUILTIN
  __builtin_amdgcn_s_wait_asynccnt(0);
#else
  asm volatile("s_wait_asynccnt 0" ::: "memory");
#endif
}

// Stage one 25600B tile into LDS. Each wave copies a contiguous 6400B slice:
// 12 rounds of 32 lanes x 16B (512B) plus one 32 x 8B (256B) tail.
__device__ __forceinline__ void prefetch_tile(const float* __restrict__ vp,
                                              int tile, float* ldsbuf) {
  const char* g = (const char*)vp + (size_t)tile * TILE_BYTES;
  char* l = (char*)ldsbuf;
  const int wave = threadIdx.x >> 5;
  const int lane = threadIdx.x & 31;
  const int base = wave * WAVE_REGION;
#pragma unroll
  for (int r = 0; r < 12; ++r) {
    const int off = base + r * 512 + lane * 16;
    async_copy_b128(g + off, l + off);
  }
  const int off = base + 6144 + lane * 8;
  async_copy_b64(g + off, l + off);
}

// ---------------- math helpers ----------------
__device__ __forceinline__ float softplus_f(float x) {
  return fmaxf(x, 0.f) + __logf(1.f + __expf(-fabsf(x)));
}

__device__ __forceinline__ float wave_reduce(float v) {
#pragma unroll
  for (int off = 16; off > 0; off >>= 1) v += __shfl_down(v, off, 32);
  return v;  // valid in lane 0
}

// ---------------- kernels ----------------
__global__ void fs2_init_ws(float* ws) {
  if (threadIdx.x < 8) ws[threadIdx.x] = 0.f;
}

// ws[0]=lp_sum  ws[1]=pitch_sum  ws[2]=energy_sum  ws[3]=dur_sum
// ws[4]=mel_mask_count (over B*T)  ws[5]=src_mask_count (over B*S)
__global__ void fs2_small_losses(const float* __restrict__ pitch_t,
                                 const float* __restrict__ energy_t,
                                 const int* __restrict__ dur_t,
                                 const float* __restrict__ pitch_p,
                                 const float* __restrict__ energy_p,
                                 const float* __restrict__ logdur_p,
                                 const unsigned char* __restrict__ src_masks,
                                 const unsigned char* __restrict__ mel_masks,
                                 float* __restrict__ ws) {
  float psum = 0.f, esum = 0.f, dsum = 0.f, cmel = 0.f, csrc = 0.f;
  const int idx = blockIdx.x * blockDim.x + threadIdx.x;
  const int stride = gridDim.x * blockDim.x;
  for (int i = idx; i < NBT; i += stride) {
    if (!mel_masks[i]) {  // mel_mask = ~mel_masks
      cmel += 1.f;
      float d = pitch_p[i] - pitch_t[i];
      float a = fabsf(d);
      psum += (a < 1.f) ? 0.5f * d * d : a - 0.5f;
      float e = energy_p[i] - energy_t[i];
      esum += e * e;
    }
  }
  for (int i = idx; i < NBS; i += stride) {
    if (!src_masks[i]) {  // src_mask = ~src_masks
      csrc += 1.f;
      float ld = __logf((float)dur_t[i] + 1.f);
      float d = logdur_p[i] - ld;
      dsum += d * d;
    }
  }
  psum = wave_reduce(psum);
  esum = wave_reduce(esum);
  dsum = wave_reduce(dsum);
  cmel = wave_reduce(cmel);
  csrc = wave_reduce(csrc);
  if ((threadIdx.x & 31) == 0) {
    atomicAdd(&ws[1], psum);
    atomicAdd(&ws[2], esum);
    atomicAdd(&ws[3], dsum);
    atomicAdd(&ws[4], cmel);
    atomicAdd(&ws[5], csrc);
  }
}

// Streaming TV-CGMM NLL: double-buffered async-to-LDS over 221MB of predictions.
__global__ void __launch_bounds__(VL_THREADS)
fs2_value_loss(const float* __restrict__ vp,
               const float* __restrict__ mel,
               const unsigned char* __restrict__ mel_masks,
               float* __restrict__ ws) {
  __shared__ __align__(16) float lds[2][TILE_BYTES / 4];
  float lp_sum = 0.f;

  int tile = blockIdx.x;
  int bufi = 0;
  if (tile < NTILES) prefetch_tile(vp, tile, lds[0]);
  while (tile < NTILES) {
    const int next = tile + gridDim.x;
    const bool have_next = next < NTILES;
    if (have_next) {
      prefetch_tile(vp, next, lds[bufi ^ 1]);
      wait_async_prev();   // tile `tile` (issued first, in-order) has landed
    } else {
      wait_async_all();
    }
    __syncthreads();       // all waves' staging of this tile visible

    {
      const int cell = tile * TILE_CELLS + threadIdx.x;
      const int m = cell % M_;
      const int bt = cell / M_;
      const int t = bt % T_;
      if (!mel_masks[bt]) {
        const float* c = &lds[bufi][threadIdx.x * 50];
        const float x0 = mel[cell];
        const float x1 = (t < T_ - 1) ? mel[cell + M_] : 0.f;
        const float x2 = (m < M_ - 1) ? mel[cell + 1] : 0.f;
        float wv[KMIX], sv[KMIX];
        float mw = -3.0e38f, msv = -3.0e38f;
#pragma unroll
        for (int k = 0; k < KMIX; ++k) {
          const float2* p2 = (const float2*)(c + k * 10);  // 8B-aligned, bank-conflict-free
          const float2 a0 = p2[0], a1 = p2[1], a2 = p2[2], a3 = p2[3], a4 = p2[4];
          const float w = a0.x;
          const float mu0 = a0.y, mu1 = a1.x, mu2 = a1.y;
          const float d0 = softplus_f(a2.x) + MIN_VAR;
          const float d1 = softplus_f(a2.y) + MIN_VAR;
          const float d2 = softplus_f(a3.x) + MIN_VAR;
          const float l10 = a3.y, l21 = a4.x, l20 = a4.y;
          const float r0 = __builtin_amdgcn_rcpf(d0);
          const float r1 = __builtin_amdgcn_rcpf(d1);
          const float r2 = __builtin_amdgcn_rcpf(d2);
          const float y0 = (x0 - mu0) * r0;
          const float y1 = (x1 - mu1 - l10 * y0) * r1;
          const float y2 = (x2 - mu2 - l20 * y0 - l21 * y1) * r2;
          const float quad = y0 * y0 + y1 * y1 + y2 * y2;
          const float logdet = __logf(d0 * d1 * d2);
          const float comp = -0.5f * quad - logdet - 1.5f * LOG_2PI;
          wv[k] = w;
          sv[k] = w + comp;
          mw = fmaxf(mw, wv[k]);
          msv = fmaxf(msv, sv[k]);
        }
        float sumw = 0.f, sums = 0.f;
#pragma unroll
        for (int k = 0; k < KMIX; ++k) {
          sumw += __expf(wv[k] - mw);
          sums += __expf(sv[k] - msv);
        }
        // lp = LSE(w+comp) - LSE(w)   (== logsumexp(log_softmax(w)+comp))
        lp_sum += (msv + __logf(sums)) - (mw + __logf(sumw));
      }
    }
    __syncthreads();       // readers done before this buffer is re-staged
    tile = next;
    bufi ^= 1;
  }

  lp_sum = wave_reduce(lp_sum);
  if ((threadIdx.x & 31) == 0) atomicAdd(&ws[0], lp_sum);
}

__global__ void fs2_finalize(const float* __restrict__ ws, float* __restrict__ out) {
  if (threadIdx.x == 0 && blockIdx.x == 0) {
    const float cmel = ws[4];
    const float csrc = ws[5];
    const float value_loss = -ws[0] / (cmel * (float)M_);
    const float pitch_loss = ws[1] / cmel;
    const float energy_loss = ws[2] / cmel;
    const float duration_loss = ws[3] / csrc;
    out[0] = value_loss + duration_loss + pitch_loss + energy_loss;
    out[1] = value_loss;
    out[2] = pitch_loss;
    out[3] = energy_loss;
    out[4] = duration_loss;
  }
}

// ---------------- launcher ----------------
extern "C" void kernel_launch(void* const* d_in, const int* in_sizes, int n_in,
                              void* d_out, int out_size, void* d_ws, size_t ws_size,
                              hipStream_t stream) {
  const float* mel_targets       = (const float*)d_in[0];
  const float* pitch_targets     = (const float*)d_in[1];
  const float* energy_targets    = (const float*)d_in[2];
  const int*   duration_targets  = (const int*)d_in[3];
  const float* value_predictions = (const float*)d_in[4];
  const float* pitch_predictions = (const float*)d_in[5];
  const float* energy_predictions= (const float*)d_in[6];
  const float* log_duration_pred = (const float*)d_in[7];
  const unsigned char* src_masks = (const unsigned char*)d_in[8];
  const unsigned char* mel_masks = (const unsigned char*)d_in[9];
  float* out = (float*)d_out;
  float* ws  = (float*)d_ws;

  fs2_init_ws<<<1, 32, 0, stream>>>(ws);
  fs2_small_losses<<<54, 256, 0, stream>>>(pitch_targets, energy_targets,
                                           duration_targets, pitch_predictions,
                                           energy_predictions, log_duration_pred,
                                           src_masks, mel_masks, ws);
  fs2_value_loss<<<VL_BLOCKS, VL_THREADS, 0, stream>>>(value_predictions,
                                                       mel_targets, mel_masks, ws);
  fs2_finalize<<<1, 32, 0, stream>>>(ws, out);
}